// DTW_23948737642967
// MI455X (gfx1250) — compile-verified
//
#include <hip/hip_runtime.h>

// DTW on MI455X (gfx1250, wave32):
//   Phase 1: row norms of ref/tgt (wave-per-row, shuffle reduce)
//   Phase 2: dist[T,S] = sqrt(max(r2+t2-2*ref@tgt^T, 0)) via V_WMMA_F32_16X16X4_F32
//   Phase 3: DTW DP as a register-resident systolic wavefront:
//            512 lanes x 8-column strips, shuffle handoff in-wave,
//            LDS ring + 32-step super-steps across wave boundaries.

typedef __attribute__((ext_vector_type(2))) float v2f;
typedef __attribute__((ext_vector_type(8))) float v8f;

#define TT 4096
#define SS 4096
#define DD 128
#define LDS_STRIDE 36   // 32 K-columns + pad; stride 36 dwords -> conflict-free & 16B aligned
#define BIGF 3.0e37f

// ---- DP geometry ----
#define W0 8            // columns per lane strip
#define NLANES 512      // 4096 / W0 strips  -> 16 waves, one workgroup
#define NWAVES 16
#define CH 32           // steps per super-step (barrier period)
#define DSTAG 63        // wave-to-wave stagger = CH + 31  (cross-wave gap == CH, barrier-ordered)
#define RB 128          // LDS ring depth per wave boundary (overwrite >= 63 steps after read)
// last step needed: 4095 + 31 + DSTAG*15 = 5071 -> 159 super-steps
#define SUPERS 159

#if defined(__has_builtin)
#if __has_builtin(__builtin_amdgcn_wmma_f32_16x16x4_f32)
#define HAVE_WMMA_F32X4 1
#endif
#endif

// ---------------------------------------------------------------- norms
__global__ __launch_bounds__(256) void dtw_norms_kernel(const float* __restrict__ ref,
                                                        const float* __restrict__ tgt,
                                                        float* __restrict__ norms) {
    const int wave = blockIdx.x * 8 + (threadIdx.x >> 5);   // 0..8191
    const int lane = threadIdx.x & 31;
    const float* p = (wave < TT) ? (ref + (size_t)wave * DD)
                                 : (tgt + (size_t)(wave - TT) * DD);
    float s = 0.0f;
#pragma unroll
    for (int q = 0; q < 4; ++q) {
        float x = p[lane + 32 * q];
        s += x * x;
    }
#pragma unroll
    for (int off = 16; off > 0; off >>= 1)
        s += __shfl_down(s, off, 32);
    if (lane == 0) norms[wave] = s;
}

// ---------------------------------------------------------------- dist (WMMA GEMM + epilogue)
// Block = 256 threads (8 waves). Each block computes a 128x128 tile of dist.
// Wave w: wm = w>>1 owns M rows [wm*32, wm*32+32), wn = w&1 owns N cols [wn*64, wn*64+64)
__global__ __launch_bounds__(256) void dtw_dist_kernel(const float* __restrict__ ref,
                                                       const float* __restrict__ tgt,
                                                       const float* __restrict__ norms,
                                                       float* __restrict__ dist) {
    __shared__ float la[128 * LDS_STRIDE];
    __shared__ float lb[128 * LDS_STRIDE];

    const int tileN = blockIdx.x * 128;
    const int tileM = blockIdx.y * 128;
    const int tid  = threadIdx.x;
    const int lane = tid & 31;
    const int w    = tid >> 5;
    const int wm   = w >> 1;   // 0..3
    const int wn   = w & 1;    // 0..1

    v8f c[2][4];
#pragma unroll
    for (int tm = 0; tm < 2; ++tm)
#pragma unroll
        for (int tn = 0; tn < 4; ++tn)
            c[tm][tn] = (v8f){0.f, 0.f, 0.f, 0.f, 0.f, 0.f, 0.f, 0.f};

    for (int kk = 0; kk < DD; kk += 32) {
#pragma unroll
        for (int q = 0; q < 4; ++q) {
            int s   = tid + 256 * q;      // 0..1023 float4 slots
            int row = s >> 3;             // 8 float4 per row
            int c4  = (s & 7) * 4;
            float4 va = *(const float4*)(ref + (size_t)(tileM + row) * DD + kk + c4);
            *(float4*)(&la[row * LDS_STRIDE + c4]) = va;
            float4 vb = *(const float4*)(tgt + (size_t)(tileN + row) * DD + kk + c4);
            *(float4*)(&lb[row * LDS_STRIDE + c4]) = vb;
        }
        __syncthreads();

#pragma unroll
        for (int kq = 0; kq < 8; ++kq) {
            const int k2 = kq * 4 + ((lane >> 4) << 1);
            v2f a[2], b[4];
#pragma unroll
            for (int tm = 0; tm < 2; ++tm)
                a[tm] = *(const v2f*)(&la[(wm * 32 + tm * 16 + (lane & 15)) * LDS_STRIDE + k2]);
#pragma unroll
            for (int tn = 0; tn < 4; ++tn)
                b[tn] = *(const v2f*)(&lb[(wn * 64 + tn * 16 + (lane & 15)) * LDS_STRIDE + k2]);

#pragma unroll
            for (int tm = 0; tm < 2; ++tm) {
#pragma unroll
                for (int tn = 0; tn < 4; ++tn) {
#if defined(HAVE_WMMA_F32X4)
                    c[tm][tn] = __builtin_amdgcn_wmma_f32_16x16x4_f32(
                        false, a[tm], false, b[tn], (short)0, c[tm][tn], false, false);
#else
#pragma unroll
                    for (int v = 0; v < 8; ++v) {
                        int rowm = wm * 32 + tm * 16 + ((lane >> 4) << 3) + v;
                        int coln = wn * 64 + tn * 16 + (lane & 15);
                        float acc = c[tm][tn][v];
#pragma unroll
                        for (int kx = 0; kx < 4; ++kx)
                            acc += la[rowm * LDS_STRIDE + kq * 4 + kx] *
                                   lb[coln * LDS_STRIDE + kq * 4 + kx];
                        c[tm][tn][v] = acc;
                    }
#endif
                }
            }
        }
        __syncthreads();
    }

    const float* r2 = norms;
    const float* t2 = norms + TT;
#pragma unroll
    for (int tm = 0; tm < 2; ++tm) {
        const int mBase = tileM + wm * 32 + tm * 16 + ((lane >> 4) << 3);
#pragma unroll
        for (int tn = 0; tn < 4; ++tn) {
            const int col = tileN + wn * 64 + tn * 16 + (lane & 15);
            const float t2v = t2[col];
#pragma unroll
            for (int v = 0; v < 8; ++v) {
                const int rowg = mBase + v;
                float d2 = r2[rowg] + t2v - 2.0f * c[tm][tn][v];
                d2 = fmaxf(d2, 0.0f);
                dist[(size_t)rowg * SS + col] = sqrtf(d2);
            }
        }
    }
}

// ---------------------------------------------------------------- DTW DP (systolic wavefront)
// acc[i][j] = dist[i][j] + min(acc[i-1][j], acc[i-1][j-1], acc[i][j-1])
// Lane G owns columns [G*8, G*8+8). At step t it processes row i = t - stag(G),
// stag = lane + DSTAG*wave. In-wave left/diag handoff: __shfl_up of the lane's
// last-column values. Cross-wave handoff: lane 31 writes its last-column value
// per row into ring[wave+1][row & 127]; lane 0 of the next wave reads it exactly
// one 32-step super-step later (barrier-ordered; slot reuse is 96 steps after read).
__global__ __launch_bounds__(512) void dtw_dp_kernel(const float* __restrict__ dist,
                                                     float* __restrict__ out) {
    __shared__ float ring[NWAVES][RB];
    const int tid  = threadIdx.x;
    const int wave = tid >> 5;
    const int lane = tid & 31;
    const int G    = tid;                   // strip index, 512 strips
    const int stag = lane + DSTAG * wave;
    const int c0   = G * W0;

    for (int s = tid; s < NWAVES * RB; s += NLANES)
        ((float*)ring)[s] = BIGF;

    float prow[W0];
#pragma unroll
    for (int x = 0; x < W0; ++x) prow[x] = BIGF;
    float lastc = BIGF, lastp = BIGF;

    // preload dist row for step 0 (only the stag==0 lane has a valid row then)
    float4 d0 = {0.f, 0.f, 0.f, 0.f}, d1 = {0.f, 0.f, 0.f, 0.f};
    if (stag == 0) {
        d0 = *(const float4*)(dist + c0);
        d1 = *(const float4*)(dist + c0 + 4);
    }
    __syncthreads();

    for (int ss = 0; ss < SUPERS; ++ss) {
#pragma unroll 2
        for (int u = 0; u < CH; ++u) {
            const int t = ss * CH + u;
            const int i = t - stag;

            // neighbor-strip boundary values (uniform shuffles)
            float leftIn = __shfl_up(lastc, 1, 32);
            float diagIn = __shfl_up(lastp, 1, 32);
            if (lane == 0) {
                if (wave == 0) {
                    leftIn = BIGF;
                    diagIn = BIGF;
                } else {
                    leftIn = ring[wave][i & (RB - 1)];
                    diagIn = ring[wave][(i - 1) & (RB - 1)];
                }
            }
            if (i == 0) diagIn = BIGF;      // diag row is row -1

            // software prefetch of next row's dist values
            const int inext = i + 1;
            const bool pf = (inext >= 0) && (inext < TT);
            float4 n0 = d0, n1 = d1;
            if (pf) {
                n0 = *(const float4*)(dist + (size_t)inext * SS + c0);
                n1 = *(const float4*)(dist + (size_t)inext * SS + c0 + 4);
            }

            if (i >= 0 && i < TT) {
                const float dv[W0] = {d0.x, d0.y, d0.z, d0.w, d1.x, d1.y, d1.z, d1.w};
                float left  = leftIn;
                float dprev = diagIn;
                float val   = 0.0f;
#pragma unroll
                for (int x = 0; x < W0; ++x) {
                    const float up = prow[x];
                    float best = fminf(left, fminf(up, dprev));
                    if (x == 0 && G == 0 && i == 0) best = 0.0f;   // origin cell
                    val = dv[x] + best;
                    dprev = up;          // old prow[x] is diag for column x+1
                    prow[x] = val;
                    left = val;
                }
                lastp = lastc;
                lastc = val;
                if (lane == 31 && wave < NWAVES - 1)
                    ring[wave + 1][i & (RB - 1)] = lastc;
            }

            d0 = n0;
            d1 = n1;
        }
        __syncthreads();
    }

    // strip 511's last column is col 4095; its last computed row is 4095
    if (tid == NLANES - 1) out[0] = lastc;
}

// ---------------------------------------------------------------- launch
extern "C" void kernel_launch(void* const* d_in, const int* in_sizes, int n_in,
                              void* d_out, int out_size, void* d_ws, size_t ws_size,
                              hipStream_t stream) {
    (void)in_sizes; (void)n_in; (void)out_size; (void)ws_size;
    const float* ref = (const float*)d_in[0];
    const float* tgt = (const float*)d_in[1];
    float* ws    = (float*)d_ws;
    float* norms = ws;              // 8192 floats
    float* dist  = ws + 8192;       // 4096*4096 floats = 64 MB (L2-resident)
    float* out   = (float*)d_out;

    dtw_norms_kernel<<<dim3(1024), dim3(256), 0, stream>>>(ref, tgt, norms);
    dtw_dist_kernel<<<dim3(SS / 128, TT / 128), dim3(256), 0, stream>>>(ref, tgt, norms, dist);
    dtw_dp_kernel<<<dim3(1), dim3(NLANES), 0, stream>>>(dist, out);
}